// MonarchOutProjection_24558622998773
// MI455X (gfx1250) — compile-verified
//
#include <hip/hip_runtime.h>

// CDNA5 / gfx1250 Monarch out-projection.
// Two per-block 32x32 GEMM phases over token tiles, fp32 WMMA 16x16x4,
// all intermediates in swizzled LDS. Memory-bound problem (~17 flop/byte),
// so the kernel is organized around: read x once (coalesced), compute via
// v_wmma_f32_16x16x4_f32, write out once (coalesced).

typedef __attribute__((ext_vector_type(2))) float v2f;
typedef __attribute__((ext_vector_type(8))) float v8f;

#define S        32
#define D        1024
#define TTOK     16        // tokens per workgroup tile
#define PITCH    1028      // LDS row pitch in floats (bank decorrelation)
#define NTHREADS 256       // 8 wave32

__global__ __launch_bounds__(NTHREADS)
void monarch_wmma_kernel(const float* __restrict__ x,
                         const float* __restrict__ L,
                         const float* __restrict__ R,
                         float* __restrict__ out)
{
    // xs: staged x tile (reused as output staging in phase 2)
    // h2s: intermediate h2 tile
    __shared__ float xs [TTOK * PITCH];
    __shared__ float h2s[TTOK * PITCH];

    const int tid  = threadIdx.x;
    const int lane = tid & 31;
    const int wave = tid >> 5;      // 0..7
    const int tl   = lane & 15;     // token row (A) / column n (B,D)
    const int kh   = lane >> 4;     // 0: K={0,1}, 1: K={2,3} per WMMA step

    const long long tile0 = (long long)blockIdx.x * TTOK;
    const float* __restrict__ xg = x   + tile0 * D;
    float*       __restrict__ og = out + tile0 * D;

    // ---------- stage x tile -> LDS, column-swizzled per 32-row ----------
    // element x[t][m*32+b] stored at xs[t*PITCH + m*32 + ((b+m)&31)]
    #pragma unroll
    for (int it = 0; it < (TTOK * D) / (NTHREADS * 4); ++it) {
        int flat = it * (NTHREADS * 4) + tid * 4;
        int t = flat >> 10;
        int i = flat & (D - 1);
        float4 v = *reinterpret_cast<const float4*>(xg + flat);
        int m = i >> 5;
        int b = i & 31;
        float* row = xs + t * PITCH + m * 32;
        row[(b + 0 + m) & 31] = v.x;
        row[(b + 1 + m) & 31] = v.y;
        row[(b + 2 + m) & 31] = v.z;
        row[(b + 3 + m) & 31] = v.w;
    }
    __syncthreads();

    // ---------- phase 1: h2[t][blk][n] = sum_m L[blk][n][m]*x[t][m*32+blk] ----------
    #pragma unroll
    for (int bi = 0; bi < 4; ++bi) {
        const int blk = wave * 4 + bi;
        const float* __restrict__ Lb = L + blk * (S * S);
        v8f acc0 = {}; // N = 0..15
        v8f acc1 = {}; // N = 16..31
        #pragma unroll
        for (int k0 = 0; k0 < S; k0 += 4) {
            const int ka = k0 + kh * 2;
            v2f a, b0, b1;
            // A fragment: A[t][k] = x[t][k*32+blk]  (swizzled LDS gather)
            a[0] = xs[tl * PITCH + (ka    ) * 32 + ((blk + ka    ) & 31)];
            a[1] = xs[tl * PITCH + (ka + 1) * 32 + ((blk + ka + 1) & 31)];
            // B fragment: B[k][n] = L[blk][n][k]
            b0[0] = Lb[(tl     ) * S + ka    ];
            b0[1] = Lb[(tl     ) * S + ka + 1];
            b1[0] = Lb[(tl + 16) * S + ka    ];
            b1[1] = Lb[(tl + 16) * S + ka + 1];
            acc0 = __builtin_amdgcn_wmma_f32_16x16x4_f32(false, a, false, b0,
                                                         (short)0, acc0, false, false);
            acc1 = __builtin_amdgcn_wmma_f32_16x16x4_f32(false, a, false, b1,
                                                         (short)0, acc1, false, false);
        }
        // D layout: VGPR r -> row (r + 8*kh), col tl (+16 for acc1)
        #pragma unroll
        for (int r = 0; r < 8; ++r) {
            const int t  = r + kh * 8;
            float* row = h2s + t * PITCH + blk * 32;
            row[((tl      + blk) & 31)] = acc0[r];
            row[((tl + 16 + blk) & 31)] = acc1[r];
        }
    }
    __syncthreads();

    // ---------- phase 2: out[t][n*32+blk] = sum_m R[blk][n][m]*h2[t][m][blk] ----------
    float* os = xs; // x tile is dead; reuse as output staging
    #pragma unroll
    for (int bi = 0; bi < 4; ++bi) {
        const int blk = wave * 4 + bi;
        const float* __restrict__ Rb = R + blk * (S * S);
        v8f acc0 = {};
        v8f acc1 = {};
        #pragma unroll
        for (int k0 = 0; k0 < S; k0 += 4) {
            const int ka = k0 + kh * 2;
            v2f a, b0, b1;
            // A fragment: A[t][m] = h2[t][m][blk]  (swizzled LDS gather)
            a[0] = h2s[tl * PITCH + (ka    ) * 32 + ((blk + ka    ) & 31)];
            a[1] = h2s[tl * PITCH + (ka + 1) * 32 + ((blk + ka + 1) & 31)];
            // B fragment: B[m][n] = R[blk][n][m]
            b0[0] = Rb[(tl     ) * S + ka    ];
            b0[1] = Rb[(tl     ) * S + ka + 1];
            b1[0] = Rb[(tl + 16) * S + ka    ];
            b1[1] = Rb[(tl + 16) * S + ka + 1];
            acc0 = __builtin_amdgcn_wmma_f32_16x16x4_f32(false, a, false, b0,
                                                         (short)0, acc0, false, false);
            acc1 = __builtin_amdgcn_wmma_f32_16x16x4_f32(false, a, false, b1,
                                                         (short)0, acc1, false, false);
        }
        // out element (t, n*32+blk) staged at os[t*PITCH + n*32 + ((blk+n)&31)]
        #pragma unroll
        for (int r = 0; r < 8; ++r) {
            const int t = r + kh * 8;
            float* row = os + t * PITCH;
            row[(tl      ) * 32 + ((blk + tl     ) & 31)] = acc0[r];
            row[(tl + 16) * 32 + ((blk + tl + 16) & 31)] = acc1[r];
        }
    }
    __syncthreads();

    // ---------- coalesced store of output tile ----------
    #pragma unroll
    for (int it = 0; it < (TTOK * D) / (NTHREADS * 4); ++it) {
        int flat = it * (NTHREADS * 4) + tid * 4;
        int t = flat >> 10;
        int i = flat & (D - 1);
        int n2 = i >> 5;
        int b  = i & 31;
        const float* row = os + t * PITCH + n2 * 32;
        float4 v;
        v.x = row[(b + 0 + n2) & 31];
        v.y = row[(b + 1 + n2) & 31];
        v.z = row[(b + 2 + n2) & 31];
        v.w = row[(b + 3 + n2) & 31];
        *reinterpret_cast<float4*>(og + flat) = v;
    }
}

extern "C" void kernel_launch(void* const* d_in, const int* in_sizes, int n_in,
                              void* d_out, int out_size, void* d_ws, size_t ws_size,
                              hipStream_t stream) {
    const float* x = (const float*)d_in[0];   // (8, 4096, 1024) f32
    const float* L = (const float*)d_in[1];   // (32, 32, 32) f32
    const float* R = (const float*)d_in[2];   // (32, 32, 32) f32
    float* out = (float*)d_out;               // (8, 4096, 1024) f32

    const int ntok = in_sizes[0] / D;         // 32768
    const int nblocks = ntok / TTOK;          // 2048 workgroups

    monarch_wmma_kernel<<<nblocks, NTHREADS, 0, stream>>>(x, L, R, out);
}